// MX2Quantizer_41609643163864
// MI455X (gfx1250) — compile-verified
//
#include <hip/hip_runtime.h>

// MX2 two-level fp4-e2m1 quantize->dequantize, single-pass streaming kernel.
// Bandwidth-bound: 512 MB total traffic -> ~22us floor at 23.3 TB/s.
// One wave32 per 128-element group; lane L holds elements [4L, 4L+3] (float4).
// Subgroup-32 amax: ds_swizzle xor-butterfly over {1,2,4}; group-128: +{8,16}.
// Butterfly runs on abs-value BIT PATTERNS with v_max_u32 (for nonnegative
// finite floats, unsigned bit order == float order) -> no fmax canonicalize ops.

typedef float v4f __attribute__((ext_vector_type(4)));

#define MX_EPS 1e-8f
#define GROUPS_PER_BLOCK 8   // 256 threads = 8 wave32s

// Direct ds_swizzle_b32, group-of-32 xor mode (ISA 11.2.2):
// offset[4:0]=and_mask=0x1F, offset[9:5]=or_mask=0, offset[14:10]=xor_mask.
template <int XOR_MASK>
__device__ __forceinline__ unsigned int swz_xor_maxu(unsigned int m) {
    unsigned int o = (unsigned int)__builtin_amdgcn_ds_swizzle((int)m, 0x1F | (XOR_MASK << 10));
    return m > o ? m : o;    // v_max_u32, single op per step
}

// Faithful port of mx _quantize_elemwise (e2m1: ebits=2, mbits=3, max_norm=6),
// followed by two-level dequant. min_exp = 0, so with be = biased exp of |n|:
//   be' = max(be, 127)            (clamps private_exp to >= 0; subnormals too)
//   sc  = 2^(1-e) = bits((255-be')<<23)   isc = 2^(e-1) = bits((be'-1)<<23)
//   q   = sign(n)*floor(|n|*sc + 0.5) * isc, clipped to +/-6   (exact pow2 rescale)
__device__ __forceinline__ float qdq_one(float n, float prod) {
    unsigned int be = (__float_as_uint(n) >> 23) & 0xFFu;   // biased exponent (v_bfe)
    be = be < 127u ? 127u : be;                             // v_max_u32
    const float sc  = __uint_as_float((255u - be) << 23);   // 2^(1-e), exact
    const float isc = __uint_as_float((be - 1u) << 23);     // 2^(e-1), exact
    float o = fabsf(n) * sc;                                // |src| folds into v_mul
    o = floorf(o + 0.5f);                                   // round nearest, ties away
    o = copysignf(o, n) * isc;                              // v_bfi + v_mul; n==0 -> 0
    o = fminf(fmaxf(o, -6.0f), 6.0f);                       // saturate normals
    return o * prod;                                        // q * scale_32 * scale_128
}

__global__ __launch_bounds__(256) void mx2_qdq_kernel(const float* __restrict__ x,
                                                      float* __restrict__ out,
                                                      unsigned int n_groups) {
    const unsigned int wave = threadIdx.x >> 5;
    const unsigned int lane = threadIdx.x & 31u;
    const unsigned int g = blockIdx.x * GROUPS_PER_BLOCK + wave;
    if (g >= n_groups) return;   // wave-uniform guard: EXEC stays all-1s

    const size_t base = (size_t)g * 128u + (size_t)lane * 4u;   // 16B-aligned

    // Streamed once; NT hints keep the 512MB stream from thrashing L2 (192MB)
    v4f v = __builtin_nontemporal_load((const v4f*)(x + base));

    // Lane-local amax of 4 elements, as unsigned bit patterns of |v|
    unsigned int ax = __float_as_uint(v.x) & 0x7fffffffu;
    unsigned int ay = __float_as_uint(v.y) & 0x7fffffffu;
    unsigned int az = __float_as_uint(v.z) & 0x7fffffffu;
    unsigned int aw = __float_as_uint(v.w) & 0x7fffffffu;
    unsigned int m3 = ax > ay ? ax : ay;                    // feeds v_max3_u32
    m3 = m3 > az ? m3 : az;
    unsigned int mb = m3 > aw ? m3 : aw;

    // Subgroup-32 amax: 8-lane cluster (lanes 8k..8k+7)
    mb = swz_xor_maxu<1>(mb);
    mb = swz_xor_maxu<2>(mb);
    mb = swz_xor_maxu<4>(mb);
    const float s32 = __uint_as_float(mb) + MX_EPS;

    // Group-128 amax: full wave32
    unsigned int tb = mb;
    tb = swz_xor_maxu<8>(tb);
    tb = swz_xor_maxu<16>(tb);
    const float s128 = __uint_as_float(tb) + MX_EPS;

    const float prod = s128 * s32;
    const float inv  = 1.0f / prod;   // one precise divide per 4 elements

    v4f r;
    r.x = qdq_one(v.x * inv, prod);
    r.y = qdq_one(v.y * inv, prod);
    r.z = qdq_one(v.z * inv, prod);
    r.w = qdq_one(v.w * inv, prod);

    __builtin_nontemporal_store(r, (v4f*)(out + base));
}

extern "C" void kernel_launch(void* const* d_in, const int* in_sizes, int n_in,
                              void* d_out, int out_size, void* d_ws, size_t ws_size,
                              hipStream_t stream) {
    (void)n_in; (void)out_size; (void)d_ws; (void)ws_size;
    const float* x = (const float*)d_in[0];
    float* out = (float*)d_out;

    const unsigned int n_elems = (unsigned int)in_sizes[0];    // 8192*8192
    const unsigned int n_groups = n_elems / 128u;              // 524288
    const unsigned int blocks = (n_groups + GROUPS_PER_BLOCK - 1) / GROUPS_PER_BLOCK;

    mx2_qdq_kernel<<<blocks, 256, 0, stream>>>(x, out, n_groups);
}